// q_phi1_58506044506633
// MI455X (gfx1250) — compile-verified
//
#include <hip/hip_runtime.h>
#include <hip/hip_bf16.h>

#define NN 100000
#define NE 1600000
#define FIN 128
#define HID 64
#define NCLS 16
#define THETA 0.5f
#define EPSN 1e-12f

typedef float v2f __attribute__((ext_vector_type(2)));
typedef float v4f __attribute__((ext_vector_type(4)));
typedef float v8f __attribute__((ext_vector_type(8)));

// ---------------- init: deg = 1 (self loop), loss accumulators = 0 ----------
__global__ __launch_bounds__(256) void k_init(float* __restrict__ deg,
                                              float* __restrict__ acc) {
  int g = blockIdx.x * blockDim.x + threadIdx.x;
  if (g < NN) deg[g] = 1.0f;
  if (g < 4) acc[g] = 0.0f;
}

// ---------------- deg[dst] += 1 per edge ------------------------------------
__global__ __launch_bounds__(256) void k_deg(const int* __restrict__ dst,
                                             float* __restrict__ deg) {
  int e = blockIdx.x * blockDim.x + threadIdx.x;
  if (e < NE) atomicAdd(&deg[dst[e]], 1.0f);
}

// ---------------- dinv = rsqrt(deg) (deg>0 guaranteed but keep guard) -------
__global__ __launch_bounds__(256) void k_dinv(float* __restrict__ deg) {
  int g = blockIdx.x * blockDim.x + threadIdx.x;
  if (g < NN) { float d = deg[g]; deg[g] = d > 0.0f ? rsqrtf(d) : 0.0f; }
}

// ---------------- H = (relu?)A @ W, fp32 WMMA 16x16 tiles per wave ----------
// A: [M,K] row-major, W: [K,N] row-major, H: [M,N]. M%16==0, N%16==0, K%4==0.
__global__ __launch_bounds__(256) void k_gemm_wmma(const float* __restrict__ A,
                                                   const float* __restrict__ W,
                                                   float* __restrict__ H,
                                                   int M, int K, int N,
                                                   int do_relu) {
  const int wave = threadIdx.x >> 5;
  const int lane = threadIdx.x & 31;
  const int tiles_n = N >> 4;
  const int tile = blockIdx.x * (blockDim.x >> 5) + wave;
  if (tile >= (M >> 4) * tiles_n) return;          // uniform per wave: EXEC stays full
  const int tn = tile % tiles_n;
  const int tm = tile / tiles_n;
  const int row0 = tm << 4, n0 = tn << 4;
  const int ml = lane & 15;                        // M (A) / N (B) within tile
  const int hi = lane >> 4;                        // selects K pair 0-1 vs 2-3
  const float* Arow = A + (size_t)(row0 + ml) * K;
  v8f acc = {};
  for (int k = 0; k < K; k += 4) {
    const int ka = k + (hi << 1);
    v2f a, b;
    a.x = Arow[ka];
    a.y = Arow[ka + 1];
    if (do_relu) { a.x = fmaxf(a.x, 0.0f); a.y = fmaxf(a.y, 0.0f); }
    b.x = W[(size_t)ka * N + n0 + ml];
    b.y = W[(size_t)(ka + 1) * N + n0 + ml];
    acc = __builtin_amdgcn_wmma_f32_16x16x4_f32(false, a, false, b,
                                                (short)0, acc, false, false);
  }
#pragma unroll
  for (int v = 0; v < 8; v++) {
    const int mm = row0 + v + (hi << 3);           // C/D layout: M = v + 8*hi
    H[(size_t)mm * N + n0 + ml] = acc[v];
  }
}

// ---------------- out[n,j] = bias[j] + H[n,j]*dinv[n]^2 (self loop) ---------
__global__ __launch_bounds__(256) void k_self_bias(const float* __restrict__ H,
                                                   const float* __restrict__ dinv,
                                                   const float* __restrict__ bias,
                                                   float* __restrict__ out) {
  int g = blockIdx.x * blockDim.x + threadIdx.x;
  if (g >= NN * HID) return;
  int n = g >> 6, j = g & 63;
  float di = dinv[n];
  out[g] = bias[j] + H[g] * di * di;
}

// ---------------- out[dst] += H[src] * dinv[src]*dinv[dst], 64 thr/edge -----
__global__ __launch_bounds__(256) void k_scatter(const int* __restrict__ src,
                                                 const int* __restrict__ dst,
                                                 const float* __restrict__ dinv,
                                                 const float* __restrict__ H,
                                                 float* __restrict__ out) {
  int e = blockIdx.x * (blockDim.x >> 6) + (threadIdx.x >> 6);
  int j = threadIdx.x & 63;
  if (e >= NE) return;
  int s = src[e], d = dst[e];
  float nrm = dinv[s] * dinv[d];
  atomicAdd(&out[(size_t)d * HID + j], H[(size_t)s * HID + j] * nrm);
}

// ---------------- rep = l2norm(l2norm(x)), one wave per node ----------------
__global__ __launch_bounds__(256) void k_normalize(const float* __restrict__ x,
                                                   float* __restrict__ rep) {
  int node = blockIdx.x * (blockDim.x >> 5) + (threadIdx.x >> 5);
  int lane = threadIdx.x & 31;
  if (node >= NN) return;
  const v2f* p = (const v2f*)(x + (size_t)node * HID);
  v2f v = p[lane];
  float ss = v.x * v.x + v.y * v.y;
#pragma unroll
  for (int o = 16; o > 0; o >>= 1) ss += __shfl_xor(ss, o, 32);
  float n1 = sqrtf(ss);
  float inv1 = 1.0f / fmaxf(n1, EPSN);
  float n2 = n1 * inv1;                       // norm after first normalize
  float inv2 = 1.0f / fmaxf(n2, EPSN);        // second (idempotent) normalize
  float s = inv1 * inv2;
  v.x *= s; v.y *= s;
  ((v2f*)(rep + (size_t)node * HID))[lane] = v;
}

// ---------------- y = rep @ Wy + by -----------------------------------------
__global__ __launch_bounds__(256) void k_logits(const float* __restrict__ rep,
                                                const float* __restrict__ Wy,
                                                const float* __restrict__ by,
                                                float* __restrict__ y) {
  __shared__ float sW[HID * NCLS];
  __shared__ float sb[NCLS];
  for (int i = threadIdx.x; i < HID * NCLS; i += blockDim.x) sW[i] = Wy[i];
  if (threadIdx.x < NCLS) sb[threadIdx.x] = by[threadIdx.x];
  __syncthreads();
  int g = blockIdx.x * blockDim.x + threadIdx.x;
  if (g >= NN * NCLS) return;
  int node = g >> 4, c = g & 15;
  const float* r = rep + (size_t)node * HID;
  float a = sb[c];
#pragma unroll
  for (int j = 0; j < HID; j++) a += r[j] * sW[j * NCLS + c];
  y[g] = a;
}

__device__ __forceinline__ float dot_rows(const float* a, const float* b, int len4) {
  const v4f* pa = (const v4f*)a;
  const v4f* pb = (const v4f*)b;
  float s = 0.0f;
  for (int i = 0; i < len4; i++) {
    v4f x = pa[i], y = pb[i];
    s += x.x * y.x + x.y * y.y + x.z * y.z + x.w * y.w;
  }
  return s;
}

// ---------------- neg loss: sum relu(dot)^2 over ns<nd ----------------------
__global__ __launch_bounds__(256) void k_neg(const int* __restrict__ nei,
                                             const float* __restrict__ rep,
                                             float* __restrict__ acc) {
  int e = blockIdx.x * blockDim.x + threadIdx.x;
  float ls = 0.0f, cs = 0.0f;
  if (e < NE) {
    int s = nei[e], d = nei[NE + e];
    if (s < d) {
      float w = fmaxf(dot_rows(rep + (size_t)s * HID, rep + (size_t)d * HID, HID / 4), 0.0f);
      ls = w * w; cs = 1.0f;
    }
  }
#pragma unroll
  for (int o = 16; o > 0; o >>= 1) { ls += __shfl_xor(ls, o, 32); cs += __shfl_xor(cs, o, 32); }
  if ((threadIdx.x & 31) == 0) { atomicAdd(&acc[0], ls); atomicAdd(&acc[1], cs); }
}

// ---------------- pos loss: sum (theta*fsim + (1-theta)*relu(dot) - 1)^2 ----
__global__ __launch_bounds__(256) void k_pos(const int* __restrict__ ei,
                                             const float* __restrict__ rep,
                                             const float* __restrict__ fo,
                                             float* __restrict__ acc) {
  int e = blockIdx.x * blockDim.x + threadIdx.x;
  float ls = 0.0f, cs = 0.0f;
  if (e < NE) {
    int s = ei[e], d = ei[NE + e];
    if (s < d) {
      float pw = fmaxf(dot_rows(rep + (size_t)s * HID, rep + (size_t)d * HID, HID / 4), 0.0f);
      float fs = dot_rows(fo + (size_t)s * FIN, fo + (size_t)d * FIN, FIN / 4);
      float p = fs * THETA + pw * (1.0f - THETA);
      float q = p - 1.0f;
      ls = q * q; cs = 1.0f;
    }
  }
#pragma unroll
  for (int o = 16; o > 0; o >>= 1) { ls += __shfl_xor(ls, o, 32); cs += __shfl_xor(cs, o, 32); }
  if ((threadIdx.x & 31) == 0) { atomicAdd(&acc[0], ls); atomicAdd(&acc[1], cs); }
}

__global__ void k_finalize(const float* __restrict__ acc, float* __restrict__ out) {
  float cnt = acc[1];
  out[0] = cnt > 0.0f ? acc[0] * (float)NN / cnt : 0.0f;
}

extern "C" void kernel_launch(void* const* d_in, const int* in_sizes, int n_in,
                              void* d_out, int out_size, void* d_ws, size_t ws_size,
                              hipStream_t stream) {
  const float* features = (const float*)d_in[0];
  const float* feat_ori = (const float*)d_in[1];
  const float* W1 = (const float*)d_in[2];
  const float* b1 = (const float*)d_in[3];
  const float* W2 = (const float*)d_in[4];
  const float* b2 = (const float*)d_in[5];
  const float* Wy = (const float*)d_in[6];
  const float* by = (const float*)d_in[7];
  const int* ei  = (const int*)d_in[8];   // [2, NE] flattened: src then dst
  const int* nei = (const int*)d_in[9];

  float* ws   = (float*)d_ws;
  float* dinv = ws;                       // NN floats (deg, then rsqrt in place)
  float* acc  = ws + 100096;              // 4 floats (loss sum, count)
  float* bufA = ws + 100352;              // NN*HID
  float* bufB = bufA + (size_t)NN * HID;  // NN*HID

  float* rep      = (float*)d_out;             // NN*HID
  float* loss_out = rep + (size_t)NN * HID;    // 1
  float* y        = loss_out + 1;              // NN*NCLS

  const int B = 256;
  const int g_node  = (NN + B - 1) / B;            // 391
  const int g_edge  = (NE + B - 1) / B;            // 6250
  const int g_nh    = (NN * HID) / B;              // 25000
  const int g_tiles = ((NN / 16) * (HID / 16)) / (B / 32);  // 3125
  const int g_scat  = NE / (B / 64);               // 400000
  const int g_norm  = NN / (B / 32) + 1;           // 12501
  const int g_y     = (NN * NCLS) / B;             // 6250

  // degrees + dinv (shared by both conv layers; edge weights are all 1)
  k_init<<<g_node, B, 0, stream>>>(dinv, acc);
  k_deg<<<g_edge, B, 0, stream>>>(ei + NE, dinv);
  k_dinv<<<g_node, B, 0, stream>>>(dinv);

  // layer 1: h1 = X @ W1 ; x1acc = b1 + selfloop + scatter
  k_gemm_wmma<<<g_tiles, B, 0, stream>>>(features, W1, bufA, NN, FIN, HID, 0);
  k_self_bias<<<g_nh, B, 0, stream>>>(bufA, dinv, b1, bufB);
  k_scatter<<<g_scat, B, 0, stream>>>(ei, ei + NE, dinv, bufA, bufB);

  // layer 2: h2 = relu(x1acc) @ W2 ; x2acc = b2 + selfloop + scatter
  k_gemm_wmma<<<g_tiles, B, 0, stream>>>(bufB, W2, bufA, NN, HID, HID, 1);
  k_self_bias<<<g_nh, B, 0, stream>>>(bufA, dinv, b2, bufB);
  k_scatter<<<g_scat, B, 0, stream>>>(ei, ei + NE, dinv, bufA, bufB);

  // rep = l2norm(l2norm(x2)) ; y = rep @ Wy + by
  k_normalize<<<g_norm, B, 0, stream>>>(bufB, rep);
  k_logits<<<g_y, B, 0, stream>>>(rep, Wy, by, y);

  // reconstruction loss
  k_neg<<<g_edge, B, 0, stream>>>(nei, rep, acc);
  k_pos<<<g_edge, B, 0, stream>>>(ei, rep, feat_ori, acc);
  k_finalize<<<1, 1, 0, stream>>>(acc, loss_out);
}